// DifferentiableStackMemory_24902220382508
// MI455X (gfx1250) — compile-verified
//
#include <hip/hip_runtime.h>

#define DIM     128
#define STACK   16
#define BATCH   32768
#define NB      2            // batch elements per workgroup
#define NTHREADS 256         // 8 wave32
#define ROWS    (NB*STACK)   // 32 rows of mem_new per WG
#define SROW    132          // padded LDS stride for 128-col input rows
#define OROW    260          // padded LDS stride for 256-col output rows
#define WK      64           // K-rows of weights staged per half
#define WROW    132
#define WMAT    (WK*WROW)    // 8448 floats per staged matrix

// ---- LDS layout (float offsets) ----
#define OFF_MR   0
#define OFF_MI   (OFF_MR + ROWS*SROW)        // 4224
#define OFF_W    (OFF_MI + ROWS*SROW)        // 8448
#define OFF_Q    (OFF_W + 2*WMAT)            // 25344
#define OFF_K    (OFF_Q + ROWS*OROW)         // 33664
#define OFF_V    (OFF_K + ROWS*OROW)         // 41984
#define OFF_PTR  (OFF_V + ROWS*OROW)         // 50304
#define OFF_AW   (OFF_PTR + ROWS)            // 50336
#define OFF_WV   (OFF_AW + NB*STACK*STACK)   // 50848
#define OFF_CNT  (OFF_WV + ROWS)             // 50880
#define LDS_FLOATS (OFF_CNT + 4)             // 50884 floats ~= 203.5 KB

typedef float v2f __attribute__((ext_vector_type(2)));
typedef float v8f __attribute__((ext_vector_type(8)));

__device__ __forceinline__ v8f wmma4(v2f a, v2f b, v8f c) {
  // D = A(16x4,f32) * B(4x16,f32) + C(16x16,f32)
  return __builtin_amdgcn_wmma_f32_16x16x4_f32(
      /*neg_a=*/false, a, /*neg_b=*/false, b,
      /*c_mod=*/(short)0, c, /*reuse_a=*/false, /*reuse_b=*/false);
}

extern "C" __global__ __launch_bounds__(NTHREADS)
void dsm_main(const float* __restrict__ z_real, const float* __restrict__ z_imag,
              const float* __restrict__ mem,    const float* __restrict__ ptr,
              const float* __restrict__ ctrl,
              const float* __restrict__ Wqr, const float* __restrict__ Wqi,
              const float* __restrict__ Wkr, const float* __restrict__ Wki,
              const float* __restrict__ Wvr, const float* __restrict__ Wvi,
              float* __restrict__ out_read, float* __restrict__ out_mem,
              float* __restrict__ out_ptr,  unsigned* __restrict__ g_cnt)
{
  extern __shared__ float lds[];
  const int t    = threadIdx.x;
  const int wg   = blockIdx.x;
  const int lane = t & 31;
  const int wave = t >> 5;

  // Warm L2/WGP$ with the first weight stage while phase 1 streams mem.
  // 256 threads x 256B covers the full 64KB of each matrix.
  __builtin_prefetch(Wqr + (t << 6), 0, 0);
  __builtin_prefetch(Wqi + (t << 6), 0, 0);

  int* cnt = (int*)&lds[OFF_CNT];
  if (t == 0) *cnt = 0;
  __syncthreads();

  // ---------------- Phase 1: gates, new_ptr, mem_new (+stage to LDS) ----
  {
    const int lb = t >> 7;        // which of the 2 batch elements
    const int tl = t & 127;
    const int b  = wg*NB + lb;
    float pu = 1.f/(1.f + expf(-ctrl[b*3+0]));
    float po = 1.f/(1.f + expf(-ctrl[b*3+1]));
    float st = 1.f/(1.f + expf(-ctrl[b*3+2]));
    float tot = pu + po + st + 1e-6f;
    pu /= tot; po /= tot; st /= tot; (void)po; (void)st;

    const float zr = z_real[b*DIM + tl];
    const float zi = z_imag[b*DIM + tl];

    #pragma unroll 4
    for (int i = 0; i < 32; ++i) {
      int idx = tl + i*128;                 // 0..4095 within this batch elem
      int s = idx >> 8, c = idx & 255;      // c = tl (i even) or tl+128 (i odd)
      float zv = (i & 1) ? zi : zr;
      // touch-once stream: keep it out of the hot L2 working set
      float mv = __builtin_nontemporal_load(&mem[(size_t)(b*STACK + s)*(2*DIM) + c]);
      float mn = mv*(1.f - pu) + pu*zv;
      __builtin_nontemporal_store(mn, &out_mem[(size_t)(b*STACK + s)*(2*DIM) + c]);
      int row = lb*STACK + s;
      if (c < DIM) lds[OFF_MR + row*SROW + c]       = mn;
      else         lds[OFF_MI + row*SROW + (c-DIM)] = mn;
    }

    if (t < 32) {
      int lb2 = t >> 4, s = t & 15;
      int b2 = wg*NB + lb2;
      float pu2 = 1.f/(1.f + expf(-ctrl[b2*3+0]));
      float po2 = 1.f/(1.f + expf(-ctrl[b2*3+1]));
      float st2 = 1.f/(1.f + expf(-ctrl[b2*3+2]));
      float tot2 = pu2 + po2 + st2 + 1e-6f;
      pu2/=tot2; po2/=tot2; st2/=tot2;
      float up = ptr[b2*STACK + ((s+15)&15)];
      float dn = ptr[b2*STACK + ((s+1)&15)];
      float cu = ptr[b2*STACK + s];
      float np = pu2*up + po2*dn + st2*cu;
      lds[OFF_PTR + lb2*STACK + s] = np;
      __builtin_nontemporal_store(np, &out_ptr[b2*STACK + s]);
      if (np > 0.1f) atomicAdd(cnt, 1);
    }
  }
  __syncthreads();
  if (t == 0) atomicAdd(g_cnt, (unsigned)(*cnt));

  // ---------------- Phase 2: complex GEMMs via V_WMMA_F32_16X16X4_F32 ---
  const float* WR[3] = {Wqr, Wkr, Wvr};
  const float* WI[3] = {Wqi, Wki, Wvi};
  const int   OUT[3] = {OFF_Q, OFF_K, OFF_V};

  const int nt = wave;          // N-tile (0..7), 16 output cols each
  const int lm = lane & 15;
  const int lh = lane >> 4;     // 0/1 half-wave selector

  for (int stage = 0; stage < 3; ++stage) {
    const float* wr = WR[stage];
    const float* wi = WI[stage];
    v8f accR[2] = {};           // real accum, M-tiles 0/1
    v8f accI[2] = {};           // imag accum

    for (int h = 0; h < 2; ++h) {
      __syncthreads();          // prior readers of OFF_W are done
      // stage W^T for k in [h*64, h*64+64): lds_w[k_local][o] = W[o][k]
      #pragma unroll 4
      for (int j = 0; j < 64; ++j) {
        int e    = t + j*NTHREADS;       // 0..16383
        int matv = e >> 13;              // 0 = Wr, 1 = Wi
        int rem  = e & 8191;
        int o    = rem >> 6;             // 0..127 (output row of W)
        int cl   = rem & 63;             // local k
        const float* Wp = matv ? wi : wr;
        lds[OFF_W + matv*WMAT + cl*WROW + o] = Wp[o*DIM + h*WK + cl];
      }
      __syncthreads();

      #pragma unroll
      for (int kk = 0; kk < 16; ++kk) {
        const int acol = h*WK + kk*4 + lh*2;
        const int krow = kk*4 + lh*2;
        v2f br, bi;
        br.x = lds[OFF_W + krow*WROW     + nt*16 + lm];
        br.y = lds[OFF_W + (krow+1)*WROW + nt*16 + lm];
        bi.x = lds[OFF_W + WMAT + krow*WROW     + nt*16 + lm];
        bi.y = lds[OFF_W + WMAT + (krow+1)*WROW + nt*16 + lm];
        #pragma unroll
        for (int mt = 0; mt < 2; ++mt) {
          int rbase = (mt*16 + lm)*SROW + acol;
          v2f ar, ai;
          ar.x = lds[OFF_MR + rbase]; ar.y = lds[OFF_MR + rbase + 1];
          ai.x = lds[OFF_MI + rbase]; ai.y = lds[OFF_MI + rbase + 1];
          v2f ain = -ai;
          accR[mt] = wmma4(ar,  br, accR[mt]);   // + mr*Wr^T
          accR[mt] = wmma4(ain, bi, accR[mt]);   // - mi*Wi^T
          accI[mt] = wmma4(ar,  bi, accI[mt]);   // + mr*Wi^T
          accI[mt] = wmma4(ai,  br, accI[mt]);   // + mi*Wr^T
        }
      }
    }
    // spill tiles: row = mt*16 + j + lh*8, col = nt*16 + lm (r) / +128 (i)
    const int ob = OUT[stage];
    #pragma unroll
    for (int mt = 0; mt < 2; ++mt)
      #pragma unroll
      for (int j = 0; j < 8; ++j) {
        int row = mt*16 + j + lh*8;
        lds[ob + row*OROW + nt*16 + lm]       = accR[mt][j];
        lds[ob + row*OROW + 128 + nt*16 + lm] = accI[mt][j];
      }
  }
  __syncthreads();

  // ---------------- Phase 3: 16x16 attention in LDS ---------------------
  #pragma unroll
  for (int i = 0; i < 2; ++i) {               // 512 (lb,s,t) score pairs
    int p  = t + i*NTHREADS;
    int lb = p >> 8, rem = p & 255;
    int s  = rem >> 4, tt = rem & 15;
    const float* qp = &lds[OFF_Q + (lb*STACK + s )*OROW];
    const float* kp = &lds[OFF_K + (lb*STACK + tt)*OROW];
    float acc = 0.f;
    #pragma unroll 8
    for (int d = 0; d < 2*DIM; ++d) acc += qp[d]*kp[d];
    lds[OFF_AW + lb*256 + s*16 + tt] = acc * 0.088388347648318447f; // DIM^-0.5
  }
  __syncthreads();

  if (t < 32) {                               // stable softmax per row
    int lb = t >> 4, s = t & 15;
    float* rowp = &lds[OFF_AW + lb*256 + s*16];
    float mx = rowp[0];
    #pragma unroll
    for (int j = 1; j < 16; ++j) mx = fmaxf(mx, rowp[j]);
    float e[16]; float sm = 0.f;
    #pragma unroll
    for (int j = 0; j < 16; ++j) { e[j] = expf(rowp[j]-mx); sm += e[j]; }
    float inv = 1.f/sm;
    #pragma unroll
    for (int j = 0; j < 16; ++j) rowp[j] = e[j]*inv;
  }
  __syncthreads();

  if (t < 32) {                               // wvec[t] = sum_s ptr[s]*aw[s,t]
    int lb = t >> 4, tt = t & 15;
    float acc = 0.f;
    #pragma unroll
    for (int s = 0; s < 16; ++s)
      acc += lds[OFF_PTR + lb*STACK + s] * lds[OFF_AW + lb*256 + s*16 + tt];
    lds[OFF_WV + lb*STACK + tt] = acc;
  }
  __syncthreads();

  #pragma unroll
  for (int i = 0; i < 2; ++i) {               // read[d] = sum_t wvec[t]*v[t,d]
    int p  = t + i*NTHREADS;
    int lb = p >> 8, d = p & 255;
    float acc = 0.f;
    #pragma unroll
    for (int tt = 0; tt < 16; ++tt)
      acc += lds[OFF_WV + lb*STACK + tt] * lds[OFF_V + (lb*STACK + tt)*OROW + d];
    __builtin_nontemporal_store(acc, &out_read[(size_t)(wg*NB + lb)*(2*DIM) + d]);
  }
}

extern "C" __global__ void dsm_zero(unsigned* g) {
  if (threadIdx.x == 0 && blockIdx.x == 0) *g = 0u;
}

extern "C" __global__ void dsm_final(const unsigned* __restrict__ g,
                                     float* __restrict__ out_scalar) {
  if (threadIdx.x == 0 && blockIdx.x == 0)
    *out_scalar = (float)(*g) / (float)BATCH;
}

extern "C" void kernel_launch(void* const* d_in, const int* in_sizes, int n_in,
                              void* d_out, int out_size, void* d_ws, size_t ws_size,
                              hipStream_t stream) {
  const float* z_real = (const float*)d_in[0];
  const float* z_imag = (const float*)d_in[1];
  const float* mem    = (const float*)d_in[2];
  const float* ptrp   = (const float*)d_in[3];
  const float* ctrl   = (const float*)d_in[4];
  const float* Wqr    = (const float*)d_in[5];
  const float* Wqi    = (const float*)d_in[6];
  const float* Wkr    = (const float*)d_in[7];
  const float* Wki    = (const float*)d_in[8];
  const float* Wvr    = (const float*)d_in[9];
  const float* Wvi    = (const float*)d_in[10];

  float* out        = (float*)d_out;
  float* out_read   = out;                                  // B x 256
  float* out_mem    = out_read + (size_t)BATCH*2*DIM;       // B x 16 x 256
  float* out_ptr    = out_mem  + (size_t)BATCH*STACK*2*DIM; // B x 16
  float* out_scalar = out_ptr  + (size_t)BATCH*STACK;       // 1

  unsigned* cnt = (unsigned*)d_ws;

  dsm_zero<<<1, 1, 0, stream>>>(cnt);
  size_t shbytes = (size_t)LDS_FLOATS * sizeof(float);      // ~203.5 KB dynamic LDS
  dsm_main<<<BATCH/NB, NTHREADS, shbytes, stream>>>(
      z_real, z_imag, mem, ptrp, ctrl,
      Wqr, Wqi, Wkr, Wki, Wvr, Wvi,
      out_read, out_mem, out_ptr, cnt);
  dsm_final<<<1, 1, 0, stream>>>(cnt, out_scalar);
}